// WassersteinCT_mat_43121471652431
// MI455X (gfx1250) — compile-verified
//
#include <hip/hip_runtime.h>
#include <hip/hip_bf16.h>

// result = 2*(C-1)/(C*C*H*W) * sum(|ct|); pred_stage1 and target are
// mathematically irrelevant (see analysis: the |one-hot*ct - one-hot*ct|
// pair-sum is 2(C-1)|ct_p| regardless of labels).

typedef __attribute__((ext_vector_type(2))) float v2f;
typedef __attribute__((ext_vector_type(8))) float v8f;

#define BLOCKS 64
#define THREADS 256

// Kernel 1: per-block partial sums of |ct|, wave-reduced via
// V_WMMA_F32_16X16X4_F32 with an all-ones B matrix (D = A x 1 = row sums).
__global__ void __launch_bounds__(THREADS)
wct_abs_partial(const float* __restrict__ ct, float* __restrict__ partial, int n) {
    const int tid    = blockIdx.x * blockDim.x + threadIdx.x;
    const int stride = gridDim.x * blockDim.x;
    const int n4     = n >> 2;

    float s = 0.0f;
    const float4* p4 = (const float4*)ct;
    for (int i = tid; i < n4; i += stride) {
        float4 v = p4[i];
        s += fabsf(v.x) + fabsf(v.y) + fabsf(v.z) + fabsf(v.w);
    }
    // tail (n not multiple of 4) -- handled by one thread, reconverges before WMMA
    if (tid == 0) {
        for (int i = n4 << 2; i < n; ++i) s += fabsf(ct[i]);
    }

    // ---- wave32 reduction via WMMA: A[16x4] holds one value per lane at
    // K=0 (lanes 0-15) / K=2 (lanes 16-31); B = ones -> D[m,n] = rowsum_m.
    v2f a; a.x = s;    a.y = 0.0f;
    v2f b; b.x = 1.0f; b.y = 1.0f;
    v8f c = {};
    c = __builtin_amdgcn_wmma_f32_16x16x4_f32(
        /*neg_a=*/false, a, /*neg_b=*/false, b,
        /*c_mod=*/(short)0, c, /*reuse_a=*/false, /*reuse_b=*/false);

    // lanes 0-15: sum of D rows 0-7 ; lanes 16-31: sum of D rows 8-15
    float half = c[0] + c[1] + c[2] + c[3] + c[4] + c[5] + c[6] + c[7];
    float wavesum = half + __shfl_xor(half, 16, 32);

    __shared__ float lds[THREADS / 32];
    const int lane = threadIdx.x & 31;
    const int wave = threadIdx.x >> 5;
    if (lane == 0) lds[wave] = wavesum;
    __syncthreads();
    if (threadIdx.x == 0) {
        float t = 0.0f;
        #pragma unroll
        for (int w = 0; w < THREADS / 32; ++w) t += lds[w];
        partial[blockIdx.x] = t;  // deterministic: no atomics
    }
}

// Kernel 2: fold the 64 partials, apply the closed-form scale.
__global__ void __launch_bounds__(64)
wct_finalize(const float* __restrict__ partial, float* __restrict__ out,
             int nparts, float scale) {
    __shared__ float lds[64];
    const int t = threadIdx.x;
    lds[t] = (t < nparts) ? partial[t] : 0.0f;
    __syncthreads();
    if (t == 0) {
        float s = 0.0f;
        for (int i = 0; i < nparts; ++i) s += lds[i];
        out[0] = s * scale;
    }
}

extern "C" void kernel_launch(void* const* d_in, const int* in_sizes, int n_in,
                              void* d_out, int out_size, void* d_ws, size_t ws_size,
                              hipStream_t stream) {
    // inputs: d_in[0]=pred_stage1 [1,8,512,512] f32 (unused),
    //         d_in[1]=ct [1,512,512] f32,
    //         d_in[2]=target (unused)
    const float* ct = (const float*)d_in[1];
    const int n = in_sizes[1];                       // 262144
    const int C = in_sizes[0] / in_sizes[1];         // 8

    float* partial = (float*)d_ws;                   // 64 floats of scratch
    float* out = (float*)d_out;

    const float scale = (2.0f * (float)(C - 1)) /
                        ((float)C * (float)C * (float)n);

    wct_abs_partial<<<BLOCKS, THREADS, 0, stream>>>(ct, partial, n);
    wct_finalize<<<1, 64, 0, stream>>>(partial, out, BLOCKS, scale);
}